// DBMLLoss_4810363372708
// MI455X (gfx1250) — compile-verified
//
#include <hip/hip_runtime.h>
#include <math.h>

// ---------------------------------------------------------------------------
// DBML loss, fused two-pass WMMA implementation for gfx1250 (MI455X).
//   B = 4096 rows, D = 512 features, sim = feats @ feats^T  (f32)
//   Pass 1: per-row  sum(sim), min_pos, max_neg  -> mean_ bar  (48 KB in ws)
//   Pass 2: recompute sim tiles, per-row sigma / fp / fn / validity,
//           loss scalar accumulated with atomicAdd into d_out.
// GEMM tiles via V_WMMA_F32_16X16X4_F32 (full f32 precision: the loss has
// exp(40*sim - 20), so sim error is amplified 40x -> low-precision WMMA
// flavors are numerically unacceptable here).
// ---------------------------------------------------------------------------

typedef __attribute__((ext_vector_type(2))) float v2f;
typedef __attribute__((ext_vector_type(8))) float v8f;

#define NB      4096       // batch
#define ND      512        // feature dim
#define TILES   256        // 4096/16 column tiles
#define CHUNKS  64         // column tiles processed 4 at a time
#define EPS_    1e-5f
#define MARGIN_ 0.1f
#define POS_A_  0.5f
#define POS_B_  0.5f
#define NEG_A_  0.5f
#define NEG_B_  0.025f
#define WEIGHT_ 0.1f
#define BIG_    1e9f

// Shared fragment loader for both A (16x4, M x K) and B (4x16, K x N == featsT):
//   lane m (m<16):  v = feats[base+m][k..k+1]
//   lane m+16:      v = feats[base+m][k+2..k+3]
__device__ __forceinline__ v2f load_frag(const float* __restrict__ feats,
                                         int base, int k, int lane) {
    const int m  = lane & 15;
    const int kk = k + ((lane >> 4) << 1);                 // +0 or +2
    const float2 p = *reinterpret_cast<const float2*>(
        feats + ((size_t)(base + m) << 9) + kk);           // <<9 == *512, 8B aligned
    v2f r; r.x = p.x; r.y = p.y;
    return r;
}

// ---------------------------------------------------------------------------
// Pass 1: per-row sum / min_pos / max_neg  ->  minPos[], maxNeg[], meanBar[]
// ---------------------------------------------------------------------------
__global__ __launch_bounds__(256) void dbml_pass1(
    const float* __restrict__ feats, const int* __restrict__ labels,
    float* __restrict__ minPosOut, float* __restrict__ maxNegOut,
    float* __restrict__ meanBarOut)
{
    const int tid  = threadIdx.x;
    const int lane = tid & 31;
    const int wave = tid >> 5;
    const int hf   = lane >> 4;             // 0: tile rows 0-7, 1: rows 8-15
    const int rowBase = blockIdx.x * 16;

    int lrow[8];
    #pragma unroll
    for (int r = 0; r < 8; ++r) lrow[r] = labels[rowBase + hf * 8 + r];

    float sum[8], mnp[8], mxn[8];
    #pragma unroll
    for (int r = 0; r < 8; ++r) { sum[r] = 0.f; mnp[r] = BIG_; mxn[r] = -BIG_; }

    for (int chunk = wave; chunk < CHUNKS; chunk += 8) {
        const int colBase = chunk * 64;     // 4 tiles * 16 cols
        v8f c[4];
        #pragma unroll
        for (int j = 0; j < 4; ++j) c[j] = (v8f){};

        for (int k = 0; k < ND; k += 4) {
            const v2f a = load_frag(feats, rowBase, k, lane);
            #pragma unroll
            for (int j = 0; j < 4; ++j) {
                const v2f b = load_frag(feats, colBase + j * 16, k, lane);
                c[j] = __builtin_amdgcn_wmma_f32_16x16x4_f32(
                    false, a, false, b, (short)0, c[j], false, false);
            }
        }

        #pragma unroll
        for (int j = 0; j < 4; ++j) {
            const int col  = colBase + j * 16 + (lane & 15);
            const int lcol = labels[col];
            #pragma unroll
            for (int r = 0; r < 8; ++r) {
                const float v    = c[j][r];
                const bool  same = (lrow[r] == lcol);
                sum[r] += v;
                if (same && v < 1.0f - EPS_) mnp[r] = fminf(mnp[r], v);
                if (!same)                   mxn[r] = fmaxf(mxn[r], v);
            }
        }
    }

    // reduce across the 16 lanes of each half-group
    #pragma unroll
    for (int off = 1; off < 16; off <<= 1) {
        #pragma unroll
        for (int r = 0; r < 8; ++r) {
            sum[r] += __shfl_xor(sum[r], off, 32);
            mnp[r]  = fminf(mnp[r], __shfl_xor(mnp[r], off, 32));
            mxn[r]  = fmaxf(mxn[r], __shfl_xor(mxn[r], off, 32));
        }
    }

    __shared__ float sS[8][16], sMn[8][16], sMx[8][16];
    if ((lane & 15) == 0) {                 // lanes 0 and 16
        #pragma unroll
        for (int r = 0; r < 8; ++r) {
            const int rr = hf * 8 + r;
            sS[wave][rr] = sum[r]; sMn[wave][rr] = mnp[r]; sMx[wave][rr] = mxn[r];
        }
    }
    __syncthreads();

    if (tid < 16) {
        float s = 0.f, mn = BIG_, mx = -BIG_;
        #pragma unroll
        for (int w = 0; w < 8; ++w) {
            s += sS[w][tid];
            mn = fminf(mn, sMn[w][tid]);
            mx = fmaxf(mx, sMx[w][tid]);
        }
        const int row = rowBase + tid;
        minPosOut[row]  = mn;
        maxNegOut[row]  = mx;
        meanBarOut[row] = (s * (1.0f / (float)NB) + 0.5f * (mn + mx)) * 0.5f;
    }
}

// ---------------------------------------------------------------------------
// Pass 2: recompute sim tiles; sigma / fp / fn / validity; atomicAdd loss.
// ---------------------------------------------------------------------------
__global__ __launch_bounds__(256) void dbml_pass2(
    const float* __restrict__ feats, const int* __restrict__ labels,
    const float* __restrict__ minPosIn, const float* __restrict__ maxNegIn,
    const float* __restrict__ meanBarIn, float* __restrict__ out)
{
    const int tid  = threadIdx.x;
    const int lane = tid & 31;
    const int wave = tid >> 5;
    const int hf   = lane >> 4;
    const int rowBase = blockIdx.x * 16;

    int   lrow[8];
    float mnp8[8], mxn8[8], mb8[8];
    #pragma unroll
    for (int r = 0; r < 8; ++r) {
        const int row = rowBase + hf * 8 + r;
        lrow[r] = labels[row];
        mnp8[r] = minPosIn[row];
        mxn8[r] = maxNegIn[row];
        mb8[r]  = meanBarIn[row];
    }

    float sig[8], fp[8], fn[8];
    unsigned ppAny = 0u, nmAny = 0u;
    #pragma unroll
    for (int r = 0; r < 8; ++r) { sig[r] = 0.f; fp[r] = 0.f; fn[r] = 0.f; }

    for (int chunk = wave; chunk < CHUNKS; chunk += 8) {
        const int colBase = chunk * 64;
        v8f c[4];
        #pragma unroll
        for (int j = 0; j < 4; ++j) c[j] = (v8f){};

        for (int k = 0; k < ND; k += 4) {
            const v2f a = load_frag(feats, rowBase, k, lane);
            #pragma unroll
            for (int j = 0; j < 4; ++j) {
                const v2f b = load_frag(feats, colBase + j * 16, k, lane);
                c[j] = __builtin_amdgcn_wmma_f32_16x16x4_f32(
                    false, a, false, b, (short)0, c[j], false, false);
            }
        }

        #pragma unroll
        for (int j = 0; j < 4; ++j) {
            const int col  = colBase + j * 16 + (lane & 15);
            const int lcol = labels[col];
            #pragma unroll
            for (int r = 0; r < 8; ++r) {
                const float v    = c[j][r];
                const bool  same = (lrow[r] == lcol);
                if (!same) {
                    const float d = v - mb8[r];
                    sig[r] += d * d;
                    if (v + MARGIN_ > mnp8[r]) {          // adaptive negatives
                        fn[r] += __expf((v - NEG_A_) * (1.0f / NEG_B_));
                        nmAny |= (1u << r);
                    }
                } else if (v < 1.0f - EPS_) {             // positives (no self)
                    if (v - MARGIN_ < mxn8[r]) {
                        fp[r] += __expf(-(v - POS_A_) * (1.0f / POS_B_));
                        ppAny |= (1u << r);
                    }
                }
            }
        }
    }

    #pragma unroll
    for (int off = 1; off < 16; off <<= 1) {
        #pragma unroll
        for (int r = 0; r < 8; ++r) {
            sig[r] += __shfl_xor(sig[r], off, 32);
            fp[r]  += __shfl_xor(fp[r],  off, 32);
            fn[r]  += __shfl_xor(fn[r],  off, 32);
        }
        ppAny |= (unsigned)__shfl_xor((int)ppAny, off, 32);
        nmAny |= (unsigned)__shfl_xor((int)nmAny, off, 32);
    }

    __shared__ float    sSg[8][16], sFp[8][16], sFn[8][16];
    __shared__ unsigned sPp[8][16], sNm[8][16];
    if ((lane & 15) == 0) {
        #pragma unroll
        for (int r = 0; r < 8; ++r) {
            const int rr = hf * 8 + r;
            sSg[wave][rr] = sig[r]; sFp[wave][rr] = fp[r]; sFn[wave][rr] = fn[r];
            sPp[wave][rr] = (ppAny >> r) & 1u;
            sNm[wave][rr] = (nmAny >> r) & 1u;
        }
    }
    __syncthreads();

    if (tid < 16) {
        float s = 0.f, f1 = 0.f, f2 = 0.f;
        unsigned pp = 0u, nm = 0u;
        #pragma unroll
        for (int w = 0; w < 8; ++w) {
            s  += sSg[w][tid]; f1 += sFp[w][tid]; f2 += sFn[w][tid];
            pp |= sPp[w][tid]; nm |= sNm[w][tid];
        }
        const int row = rowBase + tid;
        const bool hasPos = minPosIn[row] <  0.5f * BIG_;
        const bool hasNeg = maxNegIn[row] > -0.5f * BIG_;
        const bool valid  = hasPos && hasNeg && pp && nm;
        if (valid) {
            const float li = logf(1.0f + f1) + logf(1.0f + f2) + WEIGHT_ * s;
            atomicAdd(out, li * (1.0f / (float)NB));
        }
    }
}

__global__ void dbml_zero(float* __restrict__ out) { out[0] = 0.0f; }

// ---------------------------------------------------------------------------
extern "C" void kernel_launch(void* const* d_in, const int* in_sizes, int n_in,
                              void* d_out, int out_size, void* d_ws, size_t ws_size,
                              hipStream_t stream) {
    const float* feats  = (const float*)d_in[0];   // [4096,512] f32
    const int*   labels = (const int*)d_in[1];     // [4096] int32
    float* out = (float*)d_out;

    // workspace: 3 * 4096 floats of row statistics (48 KB)
    float* minPos  = (float*)d_ws;
    float* maxNeg  = minPos + NB;
    float* meanBar = maxNeg + NB;

    dim3 grid(NB / 16), block(256);
    dbml_pass1<<<grid, block, 0, stream>>>(feats, labels, minPos, maxNeg, meanBar);
    dbml_zero<<<1, 1, 0, stream>>>(out);
    dbml_pass2<<<grid, block, 0, stream>>>(feats, labels, minPos, maxNeg, meanBar, out);
}